// ContrastivePredictionLoss_33818572488744
// MI455X (gfx1250) — compile-verified
//
#include <hip/hip_runtime.h>

typedef __attribute__((ext_vector_type(2))) float v2f;
typedef __attribute__((ext_vector_type(8))) float v8f;

#define MARGIN            1.0f
#define NBATCH            64
#define N_PER_BATCH       (4 * 256 * 256)                 // 262144 elems per batch
#define BLOCKS_PER_BATCH  16
#define THREADS           256
#define VEC4_PER_BATCH    (N_PER_BATCH / 4)               // 65536 float4
#define VEC4_PER_BLOCK    (VEC4_PER_BATCH / BLOCKS_PER_BATCH) // 4096 float4
#define ITERS             (VEC4_PER_BLOCK / THREADS)      // 16 float4 per thread

// Exact fp32 wave(32)-wide sum via V_WMMA_F32_16X16X4_F32 with B = ones.
// A = {v, 0} per lane: D[m][n] = sum over all A elements contributed by
// lanes m and m+16, independent of n. Summing a lane's 8 C/D VGPRs gives
// sum_{m=0..7} D[m][lane] (lanes 0-15) / sum_{m=8..15} (lanes 16-31);
// adding the lane^16 partner yields the full 32-lane total in every lane.
__device__ __forceinline__ float wave_reduce_f32_wmma(float v) {
    v2f a;    a.x = v;    a.y = 0.0f;
    v2f ones; ones.x = 1.0f; ones.y = 1.0f;
    v8f c = {};
    c = __builtin_amdgcn_wmma_f32_16x16x4_f32(
            /*neg_a=*/false, a, /*neg_b=*/false, ones,
            /*c_mod=*/(short)0, c, /*reuse_a=*/false, /*reuse_b=*/false);
    float t = ((c[0] + c[1]) + (c[2] + c[3])) + ((c[4] + c[5]) + (c[6] + c[7]));
    t += __shfl_xor(t, 16, 32);   // combine lane n with lane n+16 -> wave total
    return t;
}

__global__ __launch_bounds__(THREADS)
void cpl_reduce_kernel(const float* __restrict__ pred_mean,
                       const float* __restrict__ pred_std,
                       const float* __restrict__ targets,
                       float* __restrict__ wsf) {
    const int blk   = blockIdx.x;
    const int batch = blk / BLOCKS_PER_BATCH;
    const int sub   = blk % BLOCKS_PER_BATCH;
    const int tid   = threadIdx.x;

    const long base = (long)batch * N_PER_BATCH;
    const float4* pm4 = (const float4*)(pred_mean + base);
    const float4* ps4 = (const float4*)(pred_std  + base);
    const float4* tg4 = (const float4*)(targets   + base);

    float err = 0.0f, unc = 0.0f;
    int idx = sub * VEC4_PER_BLOCK + tid;
#pragma unroll
    for (int it = 0; it < ITERS; ++it, idx += THREADS) {
        float4 m = pm4[idx];        // global_load_b128
        float4 t = tg4[idx];
        float4 s = ps4[idx];
        err += (fabsf(m.x - t.x) + fabsf(m.y - t.y)) +
               (fabsf(m.z - t.z) + fabsf(m.w - t.w));
        unc += (s.x + s.y) + (s.z + s.w);
    }

    // Wave-level reduction through the fp32 matrix pipe (EXEC all-ones here).
    float we = wave_reduce_f32_wmma(err);
    float wu = wave_reduce_f32_wmma(unc);

    __shared__ float se[THREADS / 32];
    __shared__ float su[THREADS / 32];
    const int wave = tid >> 5;
    if ((tid & 31) == 0) { se[wave] = we; su[wave] = wu; }
    __syncthreads();
    if (tid == 0) {
        float te = 0.0f, tu = 0.0f;
#pragma unroll
        for (int w = 0; w < THREADS / 32; ++w) { te += se[w]; tu += su[w]; }
        wsf[blk] = te;                                    // err partial
        wsf[NBATCH * BLOCKS_PER_BATCH + blk] = tu;        // unc partial
    }
}

__global__ __launch_bounds__(NBATCH)
void cpl_finalize_kernel(const float* __restrict__ wsf, float* __restrict__ out) {
    __shared__ float s_err[NBATCH];
    __shared__ float s_unc[NBATCH];
    __shared__ float s_red[NBATCH];

    const int i = threadIdx.x;  // 0..63
    float e = 0.0f, u = 0.0f;
#pragma unroll
    for (int k = 0; k < BLOCKS_PER_BATCH; ++k) {
        e += wsf[i * BLOCKS_PER_BATCH + k];
        u += wsf[NBATCH * BLOCKS_PER_BATCH + i * BLOCKS_PER_BATCH + k];
    }
    const float inv_n = 1.0f / (float)N_PER_BATCH;
    s_err[i] = e * inv_n;
    s_unc[i] = u * inv_n;
    __syncthreads();

    const float ei = s_err[i];
    const float ui = s_unc[i];
    float local = 0.0f;
    for (int j = i + 1; j < NBATCH; ++j) {
        float d = (ei > s_err[j]) ? (s_unc[j] - ui) : (ui - s_unc[j]);
        d += MARGIN;
        local += fmaxf(d, 0.0f);
    }
    s_red[i] = local;
    __syncthreads();

    if (i == 0) {
        float s = 0.0f;
#pragma unroll
        for (int k = 0; k < NBATCH; ++k) s += s_red[k];
        out[0] = s / (float)(NBATCH * (NBATCH - 1) / 2);  // / 2016
    }
}

extern "C" void kernel_launch(void* const* d_in, const int* in_sizes, int n_in,
                              void* d_out, int out_size, void* d_ws, size_t ws_size,
                              hipStream_t stream) {
    const float* pred_mean = (const float*)d_in[0];
    const float* pred_std  = (const float*)d_in[1];
    const float* targets   = (const float*)d_in[2];
    float* wsf = (float*)d_ws;     // 2 * 1024 floats = 8 KB, all slots overwritten
    float* out = (float*)d_out;

    cpl_reduce_kernel<<<NBATCH * BLOCKS_PER_BATCH, THREADS, 0, stream>>>(
        pred_mean, pred_std, targets, wsf);
    cpl_finalize_kernel<<<1, NBATCH, 0, stream>>>(wsf, out);
}